// EquivariantLieConvLayer_85048942395862
// MI455X (gfx1250) — compile-verified
//
#include <hip/hip_runtime.h>

// EquivariantLieConvLayer for MI455X (gfx1250, wave32).
//
// Pipeline (exact up to FP summation order):
//   1) k_zero    : zero S accumulator + pair-packed scatter matrix Mp in d_ws
//   2) k_buildM  : Mp[t/2][ck[t]][t%2] = cv[t]    (304x256 float2, L2-resident)
//   3) k_scatter : S[tgt] += features[src]        (bilinearity: fold edges first)
//   4) k_node    : per 16-node tile:
//        T1[n][t] = (a_bil*a_msg) * S[n][ci_t] * F[n][cj_t]   (LDS, 16x608)
//        agg      = T1 x M                        via v_wmma_f32_16x16x4_f32
//        T2[n][t] = (us*a_w) * agg[n][ci_t] * agg[n][cj_t]
//        out      = F + agg + T2 x M              via v_wmma_f32_16x16x4_f32
//   Both column tiles of a wave share one K-loop: A fragment loaded once,
//   B fragments are single 8B loads from the pair-packed M.

typedef float v2f __attribute__((ext_vector_type(2)));
typedef float v8f __attribute__((ext_vector_type(8)));

#define ALG_DIM 248
#define NNZ     600
#define KPAD    608      // 152 * 4 WMMA K-steps
#define NCOLS   256      // padded columns of M (248 valid)
#define TILE_N  16
#define FSTR    249      // odd stride: conflict-free column access
#define TSTR    611      // odd stride for T tile

// ---------------- kernel 0: zero S and Mp ----------------
__global__ void k_zero(float* __restrict__ S, float* __restrict__ M,
                       int nS, int nM) {
    int i = blockIdx.x * blockDim.x + threadIdx.x;
    int stride = gridDim.x * blockDim.x;
    for (int t = i; t < nS; t += stride) S[t] = 0.0f;
    for (int t = i; t < nM; t += stride) M[t] = 0.0f;
}

// -------- kernel 1: build pair-packed scatter matrix --------
// Mp viewed as float: Mp_f[ ((t/2)*NCOLS + ck[t])*2 + (t&1) ] = cv[t]
__global__ void k_buildM(const int* __restrict__ ck, const float* __restrict__ cv,
                         float* __restrict__ Mp_f) {
    int t = blockIdx.x * blockDim.x + threadIdx.x;
    if (t < NNZ) Mp_f[((t >> 1) * NCOLS + ck[t]) * 2 + (t & 1)] = cv[t];
}

// ---------------- kernel 2: S[tgt] += features[src] ----------------
__global__ void k_scatter(const float* __restrict__ F,
                          const int* __restrict__ srcI,
                          const int* __restrict__ tgtI,
                          float* __restrict__ S, int nEdges) {
    const int Q = ALG_DIM / 4;  // 62 float4 per row (row size 992B, 16B aligned)
    int gid = blockIdx.x * blockDim.x + threadIdx.x;
    int total = nEdges * Q;
    if (gid >= total) return;
    int e = gid / Q;
    int q = gid - e * Q;
    int s = srcI[e];
    int t = tgtI[e];
    const float4 v = ((const float4*)(F + (long long)s * ALG_DIM))[q];
    float* dst = S + (long long)t * ALG_DIM + q * 4;
    atomicAdd(dst + 0, v.x);
    atomicAdd(dst + 1, v.y);
    atomicAdd(dst + 2, v.z);
    atomicAdd(dst + 3, v.w);
}

// ---------------- kernel 3: per-node brackets via WMMA ----------------
__launch_bounds__(256, 1)
__global__ void k_node(const float* __restrict__ F,
                       const float* __restrict__ S,
                       const float2* __restrict__ Mp,
                       const int* __restrict__ ci,
                       const int* __restrict__ cj,
                       const float* __restrict__ aMsg,
                       const float* __restrict__ aBil,
                       const float* __restrict__ aW,
                       const float* __restrict__ uScale,
                       float* __restrict__ out) {
    __shared__ float sF[TILE_N * FSTR];   // feature tile
    __shared__ float sA[TILE_N * FSTR];   // S tile, then agg tile
    __shared__ float sT[TILE_N * TSTR];   // T (GEMM A operand), K-padded

    const int tid  = threadIdx.x;
    const int lane = tid & 31;
    const int wave = tid >> 5;            // 8 wave32s
    const int l15  = lane & 15;           // A row / B,D column within tile
    const int kh   = lane >> 4;           // half-wave selector
    const int node0 = blockIdx.x * TILE_N;
    const int n0 = wave * 16;             // this wave's two column tiles
    const int n1 = (wave + 8) * 16;

    const float sAgg = aBil[0] * aMsg[0];     // bilinear in 1st arg
    const float sUpd = uScale[0] * aW[0];     // bilinear in 2nd arg

    // ---- load tiles ----
    for (int idx = tid; idx < TILE_N * ALG_DIM; idx += 256) {
        int n = idx / ALG_DIM, d = idx - n * ALG_DIM;
        sF[n * FSTR + d] = F[(long long)(node0 + n) * ALG_DIM + d];
        sA[n * FSTR + d] = S[(long long)(node0 + n) * ALG_DIM + d];
    }
    // zero the K padding of T (t = 600..607) once; both GEMMs reuse it
    for (int idx = tid; idx < TILE_N * (KPAD - NNZ); idx += 256) {
        int n = idx / (KPAD - NNZ), t = NNZ + (idx - n * (KPAD - NNZ));
        sT[n * TSTR + t] = 0.0f;
    }
    __syncthreads();

    // ---- build T1 ----
    for (int idx = tid; idx < TILE_N * NNZ; idx += 256) {
        int n = idx / NNZ, t = idx - n * NNZ;
        sT[n * TSTR + t] = sAgg * sA[n * FSTR + ci[t]] * sF[n * FSTR + cj[t]];
    }
    __syncthreads();

    // ---- GEMM 1: agg = T1 x M (two column tiles share one K-loop) ----
    v8f d0 = {}, d1 = {};
    for (int k0 = 0; k0 < KPAD; k0 += 4) {
        // A 16x4 f32 layout: VGPR0 = K {0|2}, VGPR1 = K {1|3} per half-wave
        v2f a;
        a.x = sT[l15 * TSTR + k0 + 2 * kh + 0];
        a.y = sT[l15 * TSTR + k0 + 2 * kh + 1];
        // B fragments: one aligned 8B load each from pair-packed M
        const float2 p0 = Mp[(k0 / 2 + kh) * NCOLS + n0 + l15];
        const float2 p1 = Mp[(k0 / 2 + kh) * NCOLS + n1 + l15];
        v2f b0; b0.x = p0.x; b0.y = p0.y;
        v2f b1; b1.x = p1.x; b1.y = p1.y;
        d0 = __builtin_amdgcn_wmma_f32_16x16x4_f32(
                 false, a, false, b0, (short)0, d0, false, false);
        d1 = __builtin_amdgcn_wmma_f32_16x16x4_f32(
                 false, a, false, b1, (short)0, d1, false, false);
    }
    // sA no longer read; stash agg tile into it (no conflict with sT reads)
    {
        const int c0 = n0 + l15;
        const int c1 = n1 + l15;
        #pragma unroll
        for (int r = 0; r < 8; ++r)
            sA[(r + 8 * kh) * FSTR + c0] = d0[r];       // c0 < 128 always valid
        if (c1 < ALG_DIM) {
            #pragma unroll
            for (int r = 0; r < 8; ++r)
                sA[(r + 8 * kh) * FSTR + c1] = d1[r];
        }
    }
    __syncthreads();

    // ---- build T2 (update bracket; analytically ~0 by antisymmetry) ----
    for (int idx = tid; idx < TILE_N * NNZ; idx += 256) {
        int n = idx / NNZ, t = idx - n * NNZ;
        sT[n * TSTR + t] = sUpd * sA[n * FSTR + ci[t]] * sA[n * FSTR + cj[t]];
    }
    __syncthreads();

    // ---- GEMM 2 + epilogue: out = F + agg + T2 x M ----
    v8f e0 = {}, e1 = {};
    for (int k0 = 0; k0 < KPAD; k0 += 4) {
        v2f a;
        a.x = sT[l15 * TSTR + k0 + 2 * kh + 0];
        a.y = sT[l15 * TSTR + k0 + 2 * kh + 1];
        const float2 p0 = Mp[(k0 / 2 + kh) * NCOLS + n0 + l15];
        const float2 p1 = Mp[(k0 / 2 + kh) * NCOLS + n1 + l15];
        v2f b0; b0.x = p0.x; b0.y = p0.y;
        v2f b1; b1.x = p1.x; b1.y = p1.y;
        e0 = __builtin_amdgcn_wmma_f32_16x16x4_f32(
                 false, a, false, b0, (short)0, e0, false, false);
        e1 = __builtin_amdgcn_wmma_f32_16x16x4_f32(
                 false, a, false, b1, (short)0, e1, false, false);
    }
    {
        const int c0 = n0 + l15;
        const int c1 = n1 + l15;
        #pragma unroll
        for (int r = 0; r < 8; ++r) {
            int rr = r + 8 * kh;
            out[(long long)(node0 + rr) * ALG_DIM + c0] =
                sF[rr * FSTR + c0] + sA[rr * FSTR + c0] + e0[r];
        }
        if (c1 < ALG_DIM) {
            #pragma unroll
            for (int r = 0; r < 8; ++r) {
                int rr = r + 8 * kh;
                out[(long long)(node0 + rr) * ALG_DIM + c1] =
                    sF[rr * FSTR + c1] + sA[rr * FSTR + c1] + e1[r];
            }
        }
    }
}

extern "C" void kernel_launch(void* const* d_in, const int* in_sizes, int n_in,
                              void* d_out, int out_size, void* d_ws, size_t ws_size,
                              hipStream_t stream) {
    const float* features = (const float*)d_in[0];
    const int*   edges    = (const int*)d_in[1];   // (2, E) flat
    const int*   ci       = (const int*)d_in[2];
    const int*   cj       = (const int*)d_in[3];
    const int*   ck       = (const int*)d_in[4];
    const float* cv       = (const float*)d_in[5];
    const float* aMsg     = (const float*)d_in[6];
    const float* aBil     = (const float*)d_in[7];
    const float* aW       = (const float*)d_in[8];
    const float* uScale   = (const float*)d_in[9];
    float* out = (float*)d_out;

    const int nNodes = in_sizes[0] / ALG_DIM;      // 20000
    const int nEdges = in_sizes[1] / 2;            // 160000
    const int* srcI = edges;
    const int* tgtI = edges + nEdges;

    // workspace: S (nNodes*248 f32) | Mp (304*256 float2)  ~= 20.5 MB
    float* S = (float*)d_ws;
    float* Mp_f = S + (size_t)nNodes * ALG_DIM;    // offset 19,840,000 B (8B aligned)

    const int nS = nNodes * ALG_DIM;
    const int nM = KPAD * NCOLS;                   // same float count, pair-packed

    k_zero<<<1024, 256, 0, stream>>>(S, Mp_f, nS, nM);
    k_buildM<<<(NNZ + 255) / 256, 256, 0, stream>>>(ck, cv, Mp_f);

    const int scatterThreads = nEdges * (ALG_DIM / 4);
    k_scatter<<<(scatterThreads + 255) / 256, 256, 0, stream>>>(
        features, srcI, tgtI, S, nEdges);

    k_node<<<nNodes / TILE_N, 256, 0, stream>>>(
        features, S, (const float2*)Mp_f, ci, cj, aMsg, aBil, aW, uScale, out);
}